// ColorFeatureExtractor_58815282151853
// MI455X (gfx1250) — compile-verified
//
#include <hip/hip_runtime.h>

typedef unsigned int u32;
typedef __attribute__((ext_vector_type(4))) u32   v4u;
typedef __attribute__((ext_vector_type(8))) int   v8i;
typedef __attribute__((ext_vector_type(4))) int   v4i;
typedef __attribute__((ext_vector_type(4))) float v4f;

namespace {
constexpr int kBins    = 4096;   // NUM_COLOR_BINS^3
constexpr int kRow     = 2048;   // L
constexpr int kThreads = 256;    // 8 waves of 32
}

__global__ __launch_bounds__(kThreads)
void ColorFeatureExtractor_hist_kernel(const int* __restrict__ x,
                                       float* __restrict__ out) {
    __shared__ __align__(16) int hist[kBins];      // 16 KB
    __shared__ __align__(16) int rowbuf[kRow];     // 8 KB, filled by TDM
    __shared__ int valid_total;

    const u32 tid = threadIdx.x;
    const u32 row = blockIdx.x;

    // ---- Wave 0: kick off a TDM DMA of this row (8 KB) Global -> LDS ----
    if (tid < 32u) {
        const unsigned long long ga =
            (unsigned long long)(x + (unsigned long long)row * kRow);

        v4u g0;
        g0[0] = 1u;                                   // count=1 (valid), user mode
        g0[1] = (u32)(unsigned long long)&rowbuf[0];  // lds_addr (low 32 bits of LDS aperture addr)
        g0[2] = (u32)ga;                              // global_addr[31:0]
        g0[3] = ((u32)(ga >> 32) & 0x01FFFFFFu)       // global_addr[56:32]
                | 0x80000000u;                        // type=2 ("image")

        v8i g1;
        g1[0] = (int)(2u << 16);                      // data_size = 4 bytes
        g1[1] = (int)(((u32)kRow & 0xFFFFu) << 16);   // tensor_dim0[15:0] in bits[63:48]
        g1[2] = (int)(((u32)kRow >> 16) | (1u << 16));// tensor_dim0[31:16] | tensor_dim1=1
        g1[3] = (int)(((u32)kRow & 0xFFFFu) << 16);   // tensor_dim1[31:16]=0 | tile_dim0=2048
        g1[4] = 1;                                    // tile_dim1=1, tile_dim2=0 (unused)
        g1[5] = kRow;                                 // tensor_dim0_stride[31:0]
        g1[6] = 0;                                    // stride0[47:32]=0, stride1[15:0]=0
        g1[7] = 0;

        v4i gz4 = {0, 0, 0, 0};                       // groups 2/3: dims 2..4 unused
        v8i gz8 = {0, 0, 0, 0, 0, 0, 0, 0};           // extra group (6-arg toolchain form)
        __builtin_amdgcn_tensor_load_to_lds(g0, g1, gz4, gz4, gz8, 0);
    }

    // ---- All waves: zero the LDS histogram while the DMA is in flight ----
    #pragma unroll
    for (int i = (int)tid; i < kBins; i += kThreads) hist[i] = 0;
    if (tid == 0u) valid_total = 0;

    if (tid < 32u) __builtin_amdgcn_s_wait_tensorcnt(0);
    __syncthreads();

    // ---- Scatter phase: LDS atomic histogram ----
    int my_valid = 0;
    #pragma unroll
    for (int k = 0; k < kRow / kThreads; ++k) {
        const int v = rowbuf[tid + k * kThreads];
        if ((u32)v < (u32)kBins) {       // drops IGNORE (-1) and the sentinel bin
            atomicAdd(&hist[v], 1);      // ds_add_u32
            ++my_valid;
        }
    }
    atomicAdd(&valid_total, my_valid);
    __syncthreads();

    // ---- Normalize and stream out (NT b128 stores, write-once data) ----
    const int   total = valid_total;
    const float tf    = (float)total;
    const unsigned long long base = (unsigned long long)row * kBins;

    #pragma unroll
    for (int k = 0; k < kBins / (kThreads * 4); ++k) {
        const int idx = ((int)tid + k * kThreads) * 4;
        const v4i h = *(const v4i*)&hist[idx];
        v4f o;
        if (total > 0) {
            o[0] = (float)h[0] / tf;
            o[1] = (float)h[1] / tf;
            o[2] = (float)h[2] / tf;
            o[3] = (float)h[3] / tf;
        } else {
            o = (v4f){0.0f, 0.0f, 0.0f, 0.0f};
        }
        __builtin_nontemporal_store(o, (v4f*)(out + base + idx));
    }
}

extern "C" void kernel_launch(void* const* d_in, const int* in_sizes, int n_in,
                              void* d_out, int out_size, void* d_ws, size_t ws_size,
                              hipStream_t stream) {
    (void)n_in; (void)out_size; (void)d_ws; (void)ws_size;
    const int* x   = (const int*)d_in[0];
    float*     out = (float*)d_out;
    const int  nRows = in_sizes[0] / kRow;   // 4096
    ColorFeatureExtractor_hist_kernel<<<dim3(nRows), dim3(kThreads), 0, stream>>>(x, out);
}